// DynamicembEmbeddingCollection_82806969467412
// MI455X (gfx1250) — compile-verified
//
#include <hip/hip_runtime.h>
#include <hip/hip_bf16.h>

// Embedding gather with feature permutation [2,0,3,1]:
//   out[f, i, :] = tables[perm[f], indices[perm[f], i], :]
// F=4, D=64 (256 B/row). Memory-bound: ~425 MB total traffic -> ~18us @ 23.3 TB/s.
//
// grid.y = feature slot f (0..3)  -> perm/table/index/output bases are all
// block-uniform SALU work; no integer division anywhere.
// grid.x covers the N*16 float4 "chunks" of one feature, 4 chunks per thread.
//
// Fully branchless: tail handling via unsigned clamp to the last chunk, which
// makes over-range threads perform an idempotent duplicate of the owner
// thread's load+store (same source row/segment, same destination, same value).
// For the actual problem size (N*16 = 3,276,800 = 3200*1024) the clamp never
// fires. No EXEC-mask branches -> all 4 index loads clause and stay in flight
// together, then all 4 gathers, then all 4 NT stores (max MLP per thread).
//
// 16 consecutive lanes service one 64-float row -> each half-wave's
// global_load_b128 consumes the gathered 256 B row with 100% cacheline
// utilization (wave32). Non-temporal b128 stores keep the 210 MB write
// stream from evicting the ~190 MB hot gathered-row set out of the 192 MB L2.

typedef float f32x4 __attribute__((ext_vector_type(4)));

#define UNROLL 4

__global__ __launch_bounds__(256) void emb_gather_kernel(
    const int* __restrict__ indices,   // [F, N] int32
    const float* __restrict__ tables,  // [F, V, 64] f32
    float* __restrict__ out,           // [F, N, 64] f32
    unsigned int N,                    // B*L lookups per feature
    unsigned int V,                    // vocab per table
    unsigned int last_chunk,           // N*16 - 1
    unsigned int tstride)              // grid.x * blockDim.x (chunk stride)
{
    const unsigned int f = blockIdx.y;                 // output feature slot
    // perm = {2,0,3,1} packed as 2-bit fields: 0b01'11'00'10 = 0x72 (SALU only)
    const unsigned int p = (0x72u >> (2u * f)) & 3u;   // source feature

    const int*   __restrict__ ind  = indices + (size_t)p * N;
    const f32x4* __restrict__ tbl4 = (const f32x4*)tables + ((size_t)p * V << 4);
    f32x4*       __restrict__ out4 = (f32x4*)out + ((size_t)f * (last_chunk + 1));

    const unsigned int g = blockIdx.x * blockDim.x + threadIdx.x;

    unsigned int ck[UNROLL];
    unsigned int seg[UNROLL];
    unsigned int idx[UNROLL];

    // Phase 1: issue all index loads back-to-back (no branches -> they clause
    // and remain in flight together).
#pragma unroll
    for (int k = 0; k < UNROLL; ++k) {
        unsigned int chunk = g + (unsigned int)k * tstride;
        chunk = chunk < last_chunk ? chunk : last_chunk;   // branchless clamp
        ck[k]  = chunk;
        seg[k] = chunk & 15u;
        idx[k] = (unsigned int)ind[chunk >> 4];  // 16 lanes share addr -> 1 req
    }

    // Phase 2: issue all gathers (4 b128 loads in flight per thread).
    f32x4 v[UNROLL];
#pragma unroll
    for (int k = 0; k < UNROLL; ++k) {
        v[k] = tbl4[(((size_t)idx[k]) << 4) + seg[k]];
    }

    // Phase 3: non-temporal b128 stores (output is write-once, never re-read).
#pragma unroll
    for (int k = 0; k < UNROLL; ++k) {
        __builtin_nontemporal_store(v[k], &out4[ck[k]]);
    }
}

extern "C" void kernel_launch(void* const* d_in, const int* in_sizes, int n_in,
                              void* d_out, int out_size, void* d_ws, size_t ws_size,
                              hipStream_t stream) {
    (void)n_in; (void)d_ws; (void)ws_size; (void)out_size;

    const int*   indices = (const int*)d_in[0];    // [F, N] (int32 per harness)
    const float* tables  = (const float*)d_in[1];  // [F, V, 64]
    float*       out     = (float*)d_out;          // [F, N, 64]

    const unsigned int F = 4;
    const unsigned int D = 64;
    const unsigned int FN = (unsigned int)in_sizes[0];        // F*N
    const unsigned int N  = FN / F;
    const unsigned int V  = (unsigned int)((size_t)in_sizes[1] / ((size_t)F * D));

    const unsigned int chunks_per_f = N * (D / 4);            // N*16 float4 per feature
    const unsigned int threads = (chunks_per_f + UNROLL - 1) / UNROLL;
    const unsigned int block = 256;
    const unsigned int grid_x = (threads + block - 1) / block;

    dim3 grid(grid_x, F, 1);
    emb_gather_kernel<<<grid, block, 0, stream>>>(
        indices, tables, out, N, V, chunks_per_f - 1, grid_x * block);
}